// SHRenderer_6579889898136
// MI455X (gfx1250) — compile-verified
//
#include <hip/hip_runtime.h>
#include <stdint.h>

typedef float v2f  __attribute__((ext_vector_type(2)));
typedef float v8f  __attribute__((ext_vector_type(8)));
typedef unsigned int u32x4 __attribute__((ext_vector_type(4)));
typedef int   i32x4 __attribute__((ext_vector_type(4)));
typedef int   i32x8 __attribute__((ext_vector_type(8)));

#define NMAX     1024
#define GTILES   (NMAX / 16)                 // 64 gaussian tiles of 16
#define WS_FLOATS (NMAX*8 + NMAX + NMAX*4)   // F[1024][8] | alpha[1024] | color[1024][4] = 13312 f32 = 53248 B

__device__ __forceinline__ float sigmoidf(float x) { return 1.0f / (1.0f + __expf(-x)); }

// ---------------------------------------------------------------------------
// Kernel 1: project gaussians, bitonic-sort by depth, emit render params in
// sorted order into d_ws. One block of 1024 threads (N <= 1024; pads get
// alpha=0 so they are no-ops in the scan).
// ---------------------------------------------------------------------------
__global__ void __launch_bounds__(1024)
prep_kernel(const float* __restrict__ mean, const float* __restrict__ qvec,
            const float* __restrict__ svb,  const float* __restrict__ sh,
            const float* __restrict__ ab,   const float* __restrict__ c2w,
            float* __restrict__ ws, int N)
{
    __shared__ float zk[NMAX];
    __shared__ int   zi[NMAX];
    const int tid = threadIdx.x;

    // Rw = c2w[:3,:3]^T  ->  Rw[i][j] = c2w[j*4+i];  t_i = c2w[i*4+3]
    float z0 = 3.0e38f;
    if (tid < N) {
        float m0 = mean[tid*3+0] - c2w[3];
        float m1 = mean[tid*3+1] - c2w[7];
        float m2 = mean[tid*3+2] - c2w[11];
        float p2 = c2w[2]*m0 + c2w[6]*m1 + c2w[10]*m2;
        z0 = fmaxf(p2, 0.001f);
    }
    zk[tid] = z0;
    zi[tid] = tid;
    __syncthreads();

    // bitonic sort ascending on zk (payload zi), 1024 elements, 1 thread each
    for (int k = 2; k <= NMAX; k <<= 1) {
        for (int j = k >> 1; j > 0; j >>= 1) {
            int ixj = tid ^ j;
            if (ixj > tid) {
                bool up = ((tid & k) == 0);
                float a = zk[tid], b = zk[ixj];
                if ((a > b) == up) {
                    zk[tid] = b; zk[ixj] = a;
                    int t = zi[tid]; zi[tid] = zi[ixj]; zi[ixj] = t;
                }
            }
            __syncthreads();
        }
    }

    // sorted position s = tid holds original gaussian n
    const int n = zi[tid];
    float* F = ws;                     // [1024][8]
    float* A = ws + NMAX*8;            // [1024]
    float* C = ws + NMAX*8 + NMAX;     // [1024][4]

    if (n >= N) {  // padding: contributes nothing
        A[tid] = 0.0f;
        C[tid*4+0] = C[tid*4+1] = C[tid*4+2] = C[tid*4+3] = 0.0f;
        #pragma unroll
        for (int i = 0; i < 8; ++i) F[tid*8+i] = 0.0f;
        return;
    }

    // recompute full camera-space position for gaussian n
    float m0 = mean[n*3+0] - c2w[3];
    float m1 = mean[n*3+1] - c2w[7];
    float m2 = mean[n*3+2] - c2w[11];
    float Rw[3][3];
    #pragma unroll
    for (int i = 0; i < 3; ++i)
        #pragma unroll
        for (int j = 0; j < 3; ++j) Rw[i][j] = c2w[j*4+i];
    float px_ = Rw[0][0]*m0 + Rw[0][1]*m1 + Rw[0][2]*m2;
    float py_ = Rw[1][0]*m0 + Rw[1][1]*m1 + Rw[1][2]*m2;
    float pz_ = Rw[2][0]*m0 + Rw[2][1]*m1 + Rw[2][2]*m2;
    float z  = fmaxf(pz_, 0.001f);
    float x  = px_, y = py_;
    float iz = 1.0f / z, iz2 = iz*iz;
    float mx = x * iz, my = y * iz;

    // quaternion -> rotation
    float qw = qvec[n*4+0], qx = qvec[n*4+1], qy = qvec[n*4+2], qz = qvec[n*4+3];
    float qn = rsqrtf(qw*qw + qx*qx + qy*qy + qz*qz);
    qw *= qn; qx *= qn; qy *= qn; qz *= qn;
    float R[3][3] = {
        {1-2*(qy*qy+qz*qz), 2*(qx*qy-qw*qz),   2*(qx*qz+qw*qy)},
        {2*(qx*qy+qw*qz),   1-2*(qx*qx+qz*qz), 2*(qy*qz-qw*qx)},
        {2*(qx*qz-qw*qy),   2*(qy*qz+qw*qx),   1-2*(qx*qx+qy*qy)}};
    float sv[3] = {__expf(svb[n*3+0]), __expf(svb[n*3+1]), __expf(svb[n*3+2])};

    // cov3d = (R*diag(sv)) (R*diag(sv))^T
    float cov3[3][3];
    #pragma unroll
    for (int i = 0; i < 3; ++i)
        #pragma unroll
        for (int j = 0; j < 3; ++j) {
            float s = 0.f;
            #pragma unroll
            for (int k = 0; k < 3; ++k) s += (R[i][k]*sv[k]) * (R[j][k]*sv[k]);
            cov3[i][j] = s;
        }
    // cov_cam = Rw cov3 Rw^T
    float tmp[3][3], covc[3][3];
    #pragma unroll
    for (int i = 0; i < 3; ++i)
        #pragma unroll
        for (int k = 0; k < 3; ++k) {
            float s = 0.f;
            #pragma unroll
            for (int j = 0; j < 3; ++j) s += Rw[i][j]*cov3[j][k];
            tmp[i][k] = s;
        }
    #pragma unroll
    for (int i = 0; i < 3; ++i)
        #pragma unroll
        for (int l = 0; l < 3; ++l) {
            float s = 0.f;
            #pragma unroll
            for (int k = 0; k < 3; ++k) s += tmp[i][k]*Rw[l][k];
            covc[i][l] = s;
        }
    // cov2d = J covc J^T with J = [[iz,0,-x iz2],[0,iz,-y iz2]]
    float J0[3] = {iz, 0.f, -x*iz2};
    float J1[3] = {0.f, iz, -y*iz2};
    float u0[3], u1[3];
    #pragma unroll
    for (int i = 0; i < 3; ++i) {
        u0[i] = covc[i][0]*J0[0] + covc[i][1]*J0[1] + covc[i][2]*J0[2];
        u1[i] = covc[i][0]*J1[0] + covc[i][1]*J1[1] + covc[i][2]*J1[2];
    }
    float ca = J0[0]*u0[0] + J0[1]*u0[1] + J0[2]*u0[2] + 1e-8f;
    float cb = J0[0]*u1[0] + J0[1]*u1[1] + J0[2]*u1[2];
    float cc = J1[0]*u1[0] + J1[1]*u1[1] + J1[2]*u1[2] + 1e-8f;
    float det = ca*cc - cb*cb;
    float ia =  cc/det, ib = -cb/det, ic = ca/det;

    // rank-6 coefficients: quad(n,p) = F[n] . {1, px, py, px^2, px*py, py^2}
    F[tid*8+0] = ia*mx*mx + 2.f*ib*mx*my + ic*my*my;
    F[tid*8+1] = -2.f*(ia*mx + ib*my);
    F[tid*8+2] = -2.f*(ib*mx + ic*my);
    F[tid*8+3] = ia;
    F[tid*8+4] = 2.f*ib;
    F[tid*8+5] = ic;
    F[tid*8+6] = 0.0f;
    F[tid*8+7] = 0.0f;

    A[tid] = sigmoidf(ab[n]);
    C[tid*4+0] = sigmoidf(0.28209479177387814f * sh[n*27 + 0*9]);
    C[tid*4+1] = sigmoidf(0.28209479177387814f * sh[n*27 + 1*9]);
    C[tid*4+2] = sigmoidf(0.28209479177387814f * sh[n*27 + 2*9]);
    C[tid*4+3] = 0.0f;
}

// ---------------------------------------------------------------------------
// Kernel 2: render. 256 threads = 8 waves; each wave owns 16 pixels.
// Param blob staged to LDS via TDM tensor_load_to_lds. Per 16-gaussian tile:
// two V_WMMA_F32_16X16X4_F32 evaluate the rank-6 quad tile, then the
// transmittance scan runs low-half gaussians (M 0..7) then high-half (8..15)
// with a shfl_xor(16) hand-off of T between lane halves.
// ---------------------------------------------------------------------------
__global__ void __launch_bounds__(256)
render_kernel(const float* __restrict__ ws, const int* __restrict__ Hp,
              const int* __restrict__ Wp, float* __restrict__ out, int P)
{
    extern __shared__ float smem[];
    float* sF = smem;                  // [1024][8]
    float* sA = smem + NMAX*8;         // [1024]
    float* sC = smem + NMAX*8 + NMAX;  // [1024][4]

    const int W = *Wp, H = *Hp;

#if __has_builtin(__builtin_amdgcn_tensor_load_to_lds) && __has_builtin(__builtin_amdgcn_s_wait_tensorcnt)
    if (threadIdx.x < 32) {            // wave-uniform: only wave 0 issues the TDM op
        // Generic pointer to __shared__ = SHARED aperture | LDS byte offset in [31:0].
        uint32_t ldsOff = (uint32_t)(uintptr_t)(void*)sF;
        uint64_t ga = (uint64_t)(uintptr_t)ws;
        const uint32_t nd = (uint32_t)WS_FLOATS;   // 13312 dword elements, 1-row 2D tile
        u32x4 g0;
        g0[0] = 1u;                                        // count=1, user desc
        g0[1] = ldsOff;                                    // lds_addr (bytes)
        g0[2] = (uint32_t)ga;                              // global_addr[31:0]
        g0[3] = (uint32_t)((ga >> 32) & 0x01FFFFFFu) | (2u << 30);  // addr[56:32] | type=2
        i32x8 g1;
        g1[0] = (int)(2u << 16);                           // wg_mask=0 (not in cluster), data_size=4B
        g1[1] = (int)((nd & 0xFFFFu) << 16);               // tensor_dim0[15:0]
        g1[2] = (int)(((nd >> 16) & 0xFFFFu) | (1u << 16));// tensor_dim0 hi | tensor_dim1=1
        g1[3] = (int)((nd & 0xFFFFu) << 16);               // tile_dim0 = nd
        g1[4] = 1;                                         // tile_dim1 = 1, tile_dim2 = 0
        g1[5] = (int)nd;                                   // tensor_dim0_stride lo
        g1[6] = 0;                                         // stride hi | dim1_stride lo
        g1[7] = 0;
        i32x4 g2 = {0,0,0,0}, g3 = {0,0,0,0};              // unused (<=2D)
        i32x8 g4 = {0,0,0,0,0,0,0,0};                      // 6-arg toolchain: extra group, unused
        __builtin_amdgcn_tensor_load_to_lds(g0, g1, g2, g3, g4, 0);
        __builtin_amdgcn_s_wait_tensorcnt(0);
    }
#else
    for (int i = threadIdx.x; i < WS_FLOATS; i += blockDim.x) sF[i] = ws[i];
#endif
    __syncthreads();

    const int lane  = threadIdx.x & 31;
    const int wave  = threadIdx.x >> 5;
    const int col16 = lane & 15;       // pixel column of this lane in the 16-wide tile
    const int half  = lane >> 4;       // 0: K0/K1 + gaussians M0..7; 1: K2/K3 + M8..15

    int pix = blockIdx.x * 128 + wave * 16 + col16;
    int pgd = (pix < P) ? pix : (P - 1);
    int prow = pgd / W, pcol = pgd % W;
    float fx = (float)W, fy = (float)W;
    float cx = 0.5f * (float)W, cy = 0.5f * (float)H;
    float px = (-cx / fx) + ((float)pcol + 0.5f) / fx;
    float py = (-cy / fy) + ((float)prow + 0.5f) / fy;

    // B fragments (4x16 f32): assume lane-half h, component v holds row K=2h+v
    // (mirrors the documented A 16x4 layout). Constant across the gaussian loop.
    v2f b1, b2;
    b1.x = half ? py      : 1.0f;      // K rows {1, px, py, px^2}
    b1.y = half ? px * px : px;
    b2.x = half ? 0.0f    : px * py;   // K rows {px*py, py^2, 0, 0}
    b2.y = half ? 0.0f    : py * py;

    float T = 1.0f;
    float accR = 0.f, accG = 0.f, accB = 0.f;

    for (int t = 0; t < GTILES; ++t) {
        // A fragment (16x4 f32): lane row M = col16, VGPR v = K(2*half+v)
        const int gg = t*16 + col16;
        v2f a1, a2;
        a1.x = sF[gg*8 + 2*half + 0];
        a1.y = sF[gg*8 + 2*half + 1];
        a2.x = sF[gg*8 + 4 + 2*half + 0];
        a2.y = sF[gg*8 + 4 + 2*half + 1];

        v8f qd = {};
        qd = __builtin_amdgcn_wmma_f32_16x16x4_f32(false, a1, false, b1, (short)0, qd, false, false);
        qd = __builtin_amdgcn_wmma_f32_16x16x4_f32(false, a2, false, b2, (short)0, qd, false, false);
        // qd[r]: quad for pixel col16, gaussian M = r + 8*half of this tile

        if (half == 0) {               // gaussians t*16 + 0..7, depth order
            #pragma unroll
            for (int r = 0; r < 8; ++r) {
                int gi = t*16 + r;
                float wv = sA[gi] * __expf(-0.5f * qd[r]);
                float contrib = (T > 0.0001f) ? wv * T : 0.0f;
                accR += contrib * sC[gi*4+0];
                accG += contrib * sC[gi*4+1];
                accB += contrib * sC[gi*4+2];
                float wcl = fminf(fmaxf(wv, 0.0f), 0.9999f);
                T *= (1.0f - wcl);
            }
        }
        float Tmid = __shfl_xor(T, 16, 32);   // hand T after M0..7 to high half
        if (half == 1) {
            T = Tmid;
            #pragma unroll
            for (int r = 0; r < 8; ++r) {
                int gi = t*16 + 8 + r;
                float wv = sA[gi] * __expf(-0.5f * qd[r]);
                float contrib = (T > 0.0001f) ? wv * T : 0.0f;
                accR += contrib * sC[gi*4+0];
                accG += contrib * sC[gi*4+1];
                accB += contrib * sC[gi*4+2];
                float wcl = fminf(fmaxf(wv, 0.0f), 0.9999f);
                T *= (1.0f - wcl);
            }
        }
        float Tfin = __shfl_xor(T, 16, 32);   // high half's final T back to low half
        if (half == 0) T = Tfin;
    }

    // combine lane-half partial colors, lanes 0..15 write their pixel
    accR += __shfl_xor(accR, 16, 32);
    accG += __shfl_xor(accG, 16, 32);
    accB += __shfl_xor(accB, 16, 32);
    if (half == 0 && pix < P) {
        out[pix*3+0] = accR;
        out[pix*3+1] = accG;
        out[pix*3+2] = accB;
    }
}

// ---------------------------------------------------------------------------
extern "C" void kernel_launch(void* const* d_in, const int* in_sizes, int n_in,
                              void* d_out, int out_size, void* d_ws, size_t ws_size,
                              hipStream_t stream)
{
    const float* mean = (const float*)d_in[0];
    const float* qvec = (const float*)d_in[1];
    const float* svb  = (const float*)d_in[2];
    const float* sh   = (const float*)d_in[3];
    const float* ab   = (const float*)d_in[4];
    const float* c2w  = (const float*)d_in[5];
    const int*   Hp   = (const int*)d_in[6];
    const int*   Wp   = (const int*)d_in[7];

    int N = in_sizes[0] / 3;       // gaussians (<= 1024 assumed)
    int P = out_size / 3;          // pixels

    float* ws = (float*)d_ws;

    prep_kernel<<<1, NMAX, 0, stream>>>(mean, qvec, svb, sh, ab, c2w, ws, N);

    int blocks = (P + 127) / 128;  // 128 pixels per 256-thread block (16/wave)
    size_t shbytes = (size_t)WS_FLOATS * sizeof(float);   // 53248 B LDS
    render_kernel<<<blocks, 256, shbytes, stream>>>(ws, Hp, Wp, (float*)d_out, P);
}